// ProposalGeneratorFCOS_29996051595656
// MI455X (gfx1250) — compile-verified
//
#include <hip/hip_runtime.h>
#include <hip/hip_bf16.h>
#include <stdint.h>

// FCOS assignment + GIoU mean for MI455X (gfx1250, wave32).
// B=32 videos, A=64 gt boxes, P=16320 positions. VALU-bound selection/reduction
// op (no matmul structure -> WMMA inapplicable). CDNA5 paths used:
//   * TENSOR_LOAD_TO_LDS (TDM DMA) to stage gt_boxes[b] (512B) into LDS,
//     waited with s_wait_tensorcnt (TENSORcnt).
//   * ds_bpermute_b32 wave32 butterfly reductions (deterministic tree).
//   * LDS broadcast reads of the 64 boxes in the inner loop.

#define B_      32
#define A_      64
#define P_      16320
#define S_      16              // slices per video
#define SLICE_  (P_ / S_)       // 1020
#define TPB_    256
#define INF_    100000000.0f

typedef unsigned int u32x4_t __attribute__((ext_vector_type(4)));
typedef int          i32x4_t __attribute__((ext_vector_type(4)));
typedef int          i32x8_t __attribute__((ext_vector_type(8)));

#if defined(__has_builtin)
#  if __has_builtin(__builtin_amdgcn_tensor_load_to_lds)
#    define HAVE_TDM 1
#  endif
#endif

// Deterministic wave32 butterfly sum via ds_bpermute_b32. After the loop every
// lane holds the full 32-lane sum.
__device__ __forceinline__ float wave32_reduce_add(float v) {
  int x = __float_as_int(v);
  const int lane = (int)(threadIdx.x & 31u);
#pragma unroll
  for (int off = 16; off >= 1; off >>= 1) {
    int src = ((lane ^ off) << 2);               // byte index, wave32 uses bits [6:2]
    int y = __builtin_amdgcn_ds_bpermute(src, x);
    x = __float_as_int(__int_as_float(x) + __int_as_float(y));
  }
  return __int_as_float(x);
}

__global__ __launch_bounds__(TPB_) void fcos_partial(
    const float* __restrict__ reg_preds,   // (B,P,2)
    const float* __restrict__ positions,   // (B,P) (rows identical -> read row 0)
    const float* __restrict__ gt_boxes,    // (B,A,2)
    const float* __restrict__ mi,          // (P,2)
    float* __restrict__ ws)                // [2 * B_*S_] partial sums / counts
{
  __shared__ float sGT[2 * A_];            // 512 B tile of gt boxes for video b
  __shared__ float sRed[2 * (TPB_ / 32)];  // per-wave partials

  const int b   = (int)blockIdx.x / S_;
  const int s   = (int)blockIdx.x % S_;
  const int tid = (int)threadIdx.x;

  // ---- Stage gt_boxes[b] (128 floats) into LDS --------------------------------
#ifdef HAVE_TDM
  if (tid < 32) {                          // wave 0 issues one TDM DMA (EXEC all-1)
    const uint64_t gaddr =
        (uint64_t)(uintptr_t)(gt_boxes + (size_t)b * (2 * A_));
    const unsigned ldsoff = (unsigned)(uintptr_t)(&sGT[0]); // flat LDS addr[31:0] = LDS offset

    u32x4_t g0;
    g0[0] = 1u;                                            // count=1, no gather
    g0[1] = ldsoff;                                        // lds_addr (bytes)
    g0[2] = (unsigned)(gaddr & 0xFFFFFFFFu);               // global_addr[31:0]
    g0[3] = (unsigned)((gaddr >> 32) & 0x1FFFFFFu)         // global_addr[56:32]
          | (2u << 30);                                    // type = 2 ("image")

    i32x8_t g1;
    g1[0] = (int)(2u << 16);        // workgroup_mask=0, data_size=2 (4B elems)
    g1[1] = (int)(128u << 16);      // tensor_dim0 = 128 elements (bits 79:48)
    g1[2] = (int)(1u << 16);        // tensor_dim0 hi=0, tensor_dim1 = 1
    g1[3] = (int)(128u << 16);      // tile_dim0 = 128
    g1[4] = 1;                      // tile_dim1 = 1, tile_dim2 = 0
    g1[5] = 128;                    // tensor_dim0_stride = 128 (lo 32)
    g1[6] = (int)(128u << 16);      // stride0 hi=0, tensor_dim1_stride lo = 128
    g1[7] = 0;

    i32x4_t gz = {0, 0, 0, 0};
#if __clang_major__ >= 23
    i32x8_t gz8 = {0, 0, 0, 0, 0, 0, 0, 0};
    __builtin_amdgcn_tensor_load_to_lds(g0, g1, gz, gz, gz8, 0);
#else
    __builtin_amdgcn_tensor_load_to_lds(g0, g1, gz, gz, 0);
#endif
#if defined(__has_builtin) && __has_builtin(__builtin_amdgcn_s_wait_tensorcnt)
    __builtin_amdgcn_s_wait_tensorcnt(0);
#else
    asm volatile("s_wait_tensorcnt 0x0" ::: "memory");
#endif
  }
#else
  if (tid < 2 * A_) sGT[tid] = gt_boxes[(size_t)b * (2 * A_) + tid];
#endif
  __syncthreads();

  // ---- Per-position assignment + GIoU ----------------------------------------
  const float* rp  = reg_preds + (size_t)b * P_ * 2;
  const int pbase  = s * SLICE_;
  const int pend   = pbase + SLICE_;

  float sum = 0.0f, cnt = 0.0f;

  for (int p = pbase + tid; p < pend; p += TPB_) {
    const float  pos  = positions[p];                       // broadcast row
    const float2 lohi = ((const float2*)mi)[p];             // (lo, hi)
    const float2 pr   = ((const float2*)rp)[p];             // (pred_l, pred_r)

    float best = 3.4e38f;         // argmin over 64 finite dists; strict '<' keeps
    float l_t = 0.0f, r_t = 0.0f; // first index on ties (matches jnp.argmin)
    bool  is_pos = false;

#pragma unroll 4
    for (int a = 0; a < A_; ++a) {
      const float gs = sGT[2 * a];          // LDS broadcast, conflict-free
      const float ge = sGT[2 * a + 1];
      const float l = pos - gs;
      const float r = ge - pos;
      const float cmin = fminf(l, r);
      const float cmax = fmaxf(l, r);
      const bool valid = (cmin > 0.0f) && (cmax >= lohi.x) && (cmax < lohi.y);
      const float center = valid ? (0.5f * (l + r) + gs) : INF_;
      const float dist = fabsf(pos - center);
      if (dist < best) { best = dist; l_t = l; r_t = r; }
      is_pos |= valid;
    }

    if (!is_pos) { l_t = 0.0f; r_t = 0.0f; }
    const float ratio = fminf(l_t, r_t) / fmaxf(fmaxf(l_t, r_t), 1e-8f);
    const float cness = is_pos ? sqrtf(fmaxf(ratio, 0.0f)) : 0.0f;

    const float pmin = pos - pr.x, pmax = pos + pr.y;
    const float gmin = pos - l_t,  gmax = pos + r_t;
    const float overlap = fmaxf(fminf(pmax, gmax) - fmaxf(pmin, gmin), 0.0f);
    const float uni = fmaxf((pmax - pmin + 1.0f) + (gmax - gmin + 1.0f) - overlap,
                            1e-4f);
    const float iou = overlap / uni;
    const float enc = fmaxf(fmaxf(pmax, gmax) - fminf(pmin, gmin), 1e-4f);
    const float g = fminf(fmaxf(1.0f - iou + (enc - uni) / enc, -1.0f), 1.0f)
                  * cness;

    if (is_pos) { sum += g; cnt += 1.0f; }
  }

  // ---- Block reduction: wave32 butterfly, then 8-wave LDS combine ------------
  sum = wave32_reduce_add(sum);
  cnt = wave32_reduce_add(cnt);
  const int lane = tid & 31, wv = tid >> 5;
  if (lane == 0) { sRed[wv] = sum; sRed[(TPB_ / 32) + wv] = cnt; }
  __syncthreads();
  if (tid == 0) {
    float ts = 0.0f, tc = 0.0f;
#pragma unroll
    for (int w = 0; w < TPB_ / 32; ++w) {
      ts += sRed[w];
      tc += sRed[(TPB_ / 32) + w];
    }
    ws[(size_t)blockIdx.x]               = ts;   // fully rewritten each call
    ws[(size_t)(B_ * S_) + blockIdx.x]   = tc;
  }
}

// One wave: combine slices per video (fixed order, deterministic), apply the
// npos rule, wave-reduce 32 per-video values, write mean.
__global__ __launch_bounds__(32) void fcos_final(const float* __restrict__ ws,
                                                 float* __restrict__ out) {
  const int b = (int)threadIdx.x;         // 32 threads == one wave32
  float s = 0.0f, c = 0.0f;
#pragma unroll
  for (int i = 0; i < S_; ++i) {
    s += ws[(size_t)b * S_ + i];
    c += ws[(size_t)(B_ * S_) + (size_t)b * S_ + i];
  }
  float pv = (c > 0.0f) ? (2.0f * s / c) : 0.0f;
  pv = wave32_reduce_add(pv);
  if (b == 0) out[0] = pv * (1.0f / (float)B_);
}

extern "C" void kernel_launch(void* const* d_in, const int* in_sizes, int n_in,
                              void* d_out, int out_size, void* d_ws, size_t ws_size,
                              hipStream_t stream) {
  const float* reg_preds = (const float*)d_in[0];   // (B,P,2)
  const float* positions = (const float*)d_in[1];   // (B,P)
  const float* gt_boxes  = (const float*)d_in[2];   // (B,A,2)
  const float* mi        = (const float*)d_in[3];   // (P,2)
  float* ws = (float*)d_ws;                          // needs 2*B_*S_*4 = 4 KB

  fcos_partial<<<dim3(B_ * S_), dim3(TPB_), 0, stream>>>(
      reg_preds, positions, gt_boxes, mi, ws);
  fcos_final<<<1, 32, 0, stream>>>(ws, (float*)d_out);
}